// TFAdaptiveEmbedding_55327768707951
// MI455X (gfx1250) — compile-verified
//
#include <hip/hip_runtime.h>

// MI455X / gfx1250 adaptive-embedding forward.
// Memory-bound (~34 MB out + ~5 MB gather/proj at 23.3 TB/s); f32 WMMA
// (V_WMMA_F32_16X16X4_F32) preserves full f32 numerics of the reference.

typedef __attribute__((ext_vector_type(2))) float v2f;
typedef __attribute__((ext_vector_type(8))) float v8f;

#define D_PROJ 1024

// One bucket's masked gather-GEMM contribution for this 16-token tile.
// DEMB is compile-time (1024/256/64/16) so all K loops fully unroll.
template <int DEMB>
__device__ __forceinline__ void bucket_pass(
    int b, const float* __restrict__ emb, const float* __restrict__ proj,
    float (*sA)[68], const int* sBucket, const int* sRow,
    int m, int kq, int ln, int laneHalf, int colBase, v8f& acc)
{
    const int myB   = sBucket[m];
    const int myRow = sRow[m];
    constexpr int KC = (DEMB < 64) ? DEMB : 64;   // K-chunk staged in LDS

    for (int kbase = 0; kbase < DEMB; kbase += 64) {
        // ---- stage 16 x KC A-tile into LDS (zeros for inactive tokens) ----
        float4 v = {0.f, 0.f, 0.f, 0.f};
        if (myB == b && kq < KC) {
            v = *(const float4*)(emb + (size_t)myRow * DEMB + kbase + kq);
        }
        *(float4*)&sA[m][kq] = v;           // 16B-aligned (stride 68 floats)
        __syncthreads();

        // ---- WMMA over the chunk: D = A(16xK) * B(Kx16) + C ----
        const float* pB = proj + (size_t)kbase * D_PROJ + colBase + ln;
#pragma unroll
        for (int k0 = 0; k0 < KC; k0 += 4) {
            // A frag (ISA 7.12.2): lanes 0-15 -> K=k0,k0+1 ; lanes 16-31 -> K=k0+2,k0+3
            v2f a = *(const v2f*)&sA[ln][k0 + 2 * laneHalf];   // conflict-free b64
            // B frag 4x16: VGPR0 = rows k0 / k0+2 per lane half, VGPR1 = k0+1 / k0+3
            v2f bv;
            bv.x = pB[(size_t)(k0 + 2 * laneHalf) * D_PROJ];
            bv.y = pB[(size_t)(k0 + 2 * laneHalf + 1) * D_PROJ];
            acc = __builtin_amdgcn_wmma_f32_16x16x4_f32(
                /*neg_a=*/false, a, /*neg_b=*/false, bv,
                /*c_mod=*/(short)0, acc, /*reuse_a=*/false, /*reuse_b=*/false);
        }
        __syncthreads();
    }
}

__global__ __launch_bounds__(256) void adaptive_emb_wmma(
    const int* __restrict__ inp,
    const float* __restrict__ emb0, const float* __restrict__ proj0,
    const float* __restrict__ emb1, const float* __restrict__ proj1,
    const float* __restrict__ emb2, const float* __restrict__ proj2,
    const float* __restrict__ emb3, const float* __restrict__ proj3,
    float* __restrict__ out, int nTok)
{
    __shared__ float sA[16][68];     // 16 x 64 K-chunk, padded to 68 (bank-conflict-free)
    __shared__ int   sBucket[16];
    __shared__ int   sRow[16];

    const int tid      = threadIdx.x;
    const int tileM    = blockIdx.x;          // 16-token tile
    const int wave     = tid >> 5;            // 8 waves (wave32)
    const int lane     = tid & 31;
    const int laneHalf = lane >> 4;
    const int ln       = lane & 15;
    const int colBase  = blockIdx.y * 128 + wave * 16;

    // classify 16 tokens: bucket + in-bucket row
    if (tid < 16) {
        int t = tileM * 16 + tid;
        int bucket = -1, row = 0;
        if (t < nTok) {
            int id = inp[t];
            if      (id <  20000) { bucket = 0; row = id;          }
            else if (id <  40000) { bucket = 1; row = id -  20000; }
            else if (id < 200000) { bucket = 2; row = id -  40000; }
            else                  { bucket = 3; row = id - 200000; }
        }
        sBucket[tid] = bucket;
        sRow[tid]    = row;
    }
    __syncthreads();

    const int m  = tid >> 4;          // staging role: token row 0..15
    const int kq = (tid & 15) << 2;   // staging role: 4 consecutive K values

    v8f acc = {};
    bucket_pass<1024>(0, emb0, proj0, sA, sBucket, sRow, m, kq, ln, laneHalf, colBase, acc);
    bucket_pass< 256>(1, emb1, proj1, sA, sBucket, sRow, m, kq, ln, laneHalf, colBase, acc);
    bucket_pass<  64>(2, emb2, proj2, sA, sBucket, sRow, m, kq, ln, laneHalf, colBase, acc);
    bucket_pass<  16>(3, emb3, proj3, sA, sBucket, sRow, m, kq, ln, laneHalf, colBase, acc);

    // C/D layout: VGPR r -> row r (lanes 0-15) / row r+8 (lanes 16-31); scale = sqrt(1024)
#pragma unroll
    for (int r = 0; r < 8; ++r) {
        int row = tileM * 16 + r + 8 * laneHalf;
        if (row < nTok)
            out[(size_t)row * D_PROJ + colBase + ln] = acc[r] * 32.0f;
    }
}

extern "C" void kernel_launch(void* const* d_in, const int* in_sizes, int n_in,
                              void* d_out, int out_size, void* d_ws, size_t ws_size,
                              hipStream_t stream) {
    // setup_inputs() dict order: inp, emb_0, proj_0, emb_1, proj_1, emb_2, proj_2, emb_3, proj_3
    const int*   inp   = (const int*)  d_in[0];
    const float* emb0  = (const float*)d_in[1];
    const float* proj0 = (const float*)d_in[2];
    const float* emb1  = (const float*)d_in[3];
    const float* proj1 = (const float*)d_in[4];
    const float* emb2  = (const float*)d_in[5];
    const float* proj2 = (const float*)d_in[6];
    const float* emb3  = (const float*)d_in[7];
    const float* proj3 = (const float*)d_in[8];
    float* out = (float*)d_out;

    const int nTok = in_sizes[0];                 // 4 * 2048 = 8192
    dim3 grid((nTok + 15) / 16, D_PROJ / 128);    // 512 x 8 workgroups
    adaptive_emb_wmma<<<grid, 256, 0, stream>>>(
        inp, emb0, proj0, emb1, proj1, emb2, proj2, emb3, proj3, out, nTok);
}